// DepGcn_36584531427855
// MI455X (gfx1250) — compile-verified
//
#include <hip/hip_runtime.h>
#include <hip/hip_bf16.h>

typedef __attribute__((ext_vector_type(2))) float v2f;
typedef __attribute__((ext_vector_type(4))) float v4f;
typedef __attribute__((ext_vector_type(8))) float v8f;

#define Bsz 8
#define Nn  256
#define Dd  128   // D_IN == D_OUT == 128

// -----------------------------------------------------------------------------
// Kernel 1: u[b,i,e] = sum_j adj[b,i,j] * (hidden[b,j,e] + dep_embed[b,j,i,e])
// One block per (b,i). 256 threads = 8 waves; wave w handles j = w, w+8, ...
// Each wave's 32 lanes x float4 = exactly one contiguous 512B row of D=128 f32.
// dep_embed (268 MB, single-use, > 192 MB L2) is loaded NON-TEMPORAL so the
// stream does not evict the reused hidden/adj working set from L2; HBM traffic
// stays at the 268 MB floor (~11.5 us at 23.3 TB/s).
// -----------------------------------------------------------------------------
__global__ __launch_bounds__(256) void depgcn_reduce_kernel(
    const float* __restrict__ hidden,     // [B,N,D]
    const float* __restrict__ adj,        // [B,N,N]
    const float* __restrict__ dep_embed,  // [B,N,N,D]
    float* __restrict__ u)                // [B,N,D] (workspace)
{
    const int b = blockIdx.x / Nn;
    const int i = blockIdx.x % Nn;
    const int tid  = threadIdx.x;
    const int wave = tid >> 5;
    const int lane = tid & 31;

    __shared__ float adjs[Nn];
    __shared__ v4f   red[256];

    // stage adj[b,i,:] (contiguous 1KB) into LDS
    adjs[tid] = adj[((size_t)b * Nn + i) * Nn + tid];
    __syncthreads();

    const v4f* dep4 = (const v4f*)dep_embed;  // idx: ((b*N+j)*N+i)*32 + e4
    const v4f* hid4 = (const v4f*)hidden;     // idx: (b*N+j)*32 + e4

    v4f acc = {};

#pragma unroll 4
    for (int j = wave; j < Nn; j += 8) {
        const float a = adjs[j];
        // single-use stream: bypass temporal caching (th:TH_LOAD_NT)
        const v4f d = __builtin_nontemporal_load(
            &dep4[((size_t)(b * Nn + j) * Nn + i) * 32 + lane]);
        // reused across blocks: normal temporal load, stays in L2
        const v4f h = hid4[((size_t)(b * Nn + j)) * 32 + lane];
        acc.x = fmaf(a, d.x + h.x, acc.x);
        acc.y = fmaf(a, d.y + h.y, acc.y);
        acc.z = fmaf(a, d.z + h.z, acc.z);
        acc.w = fmaf(a, d.w + h.w, acc.w);
    }

    red[tid] = acc;
    __syncthreads();

    // cross-wave reduction: wave 0 sums the 8 partials per lane
    if (tid < 32) {
        v4f s = red[tid];
#pragma unroll
        for (int w = 1; w < 8; ++w) {
            const v4f p = red[w * 32 + tid];
            s.x += p.x; s.y += p.y; s.z += p.z; s.w += p.w;
        }
        v4f* u4 = (v4f*)u;
        u4[((size_t)(b * Nn + i)) * 32 + tid] = s;   // consumed by stage 2: keep RT
    }
}

// -----------------------------------------------------------------------------
// Kernel 2: out[m,d] = u[m,:] @ W[:,d] + bias[d], M = B*N = 2048, K = D = 128.
// One wave per 16x16 output tile, K swept with v_wmma_f32_16x16x4_f32 (x32).
// Fragment layouts per CDNA5 ISA 7.12.2:
//   A 16x4 f32:  lanes 0-15 hold {K=0,K=1}, lanes 16-31 hold {K=2,K=3}
//   B 4x16 f32:  VGPR holds one K-row striped across lanes (halves by K pair)
//   C/D 16x16:   VGPR r -> M=r (lanes 0-15) / M=r+8 (lanes 16-31), N = lane%16
// Uniform control flow -> EXEC all-1s as WMMA requires.
// -----------------------------------------------------------------------------
__global__ __launch_bounds__(256) void depgcn_wmma_gemm_kernel(
    const float* __restrict__ u,     // [2048,128]
    const float* __restrict__ W,     // [128,128]
    const float* __restrict__ bias,  // [128]
    float* __restrict__ out)         // [2048,128]
{
    const int wave = threadIdx.x >> 5;
    const int lane = threadIdx.x & 31;
    const int tile = blockIdx.x * 8 + wave;   // 1024 tiles total
    const int mt = tile >> 3;                 // 0..127
    const int nt = tile & 7;                  // 0..7
    const int m0 = mt * 16;
    const int n0 = nt * 16;

    const int r  = lane & 15;
    const int hi = lane >> 4;

    v8f acc = {};

#pragma unroll 8
    for (int k0 = 0; k0 < 128; k0 += 4) {
        const int kk = k0 + hi * 2;
        v2f a, bv;
        a.x  = u[(size_t)(m0 + r) * 128 + kk];
        a.y  = u[(size_t)(m0 + r) * 128 + kk + 1];
        bv.x = W[(size_t)kk * 128 + n0 + r];
        bv.y = W[(size_t)(kk + 1) * 128 + n0 + r];
        acc = __builtin_amdgcn_wmma_f32_16x16x4_f32(
            /*neg_a=*/false, a, /*neg_b=*/false, bv,
            /*c_mod=*/(short)0, acc, /*reuse_a=*/false, /*reuse_b=*/false);
    }

    const float bb  = bias[n0 + r];
    const int   col = n0 + r;
    const int   mb  = m0 + hi * 8;
#pragma unroll
    for (int vr = 0; vr < 8; ++vr) {
        out[(size_t)(mb + vr) * 128 + col] = acc[vr] + bb;
    }
}

extern "C" void kernel_launch(void* const* d_in, const int* in_sizes, int n_in,
                              void* d_out, int out_size, void* d_ws, size_t ws_size,
                              hipStream_t stream) {
    const float* hidden    = (const float*)d_in[0];  // [8,256,128]
    const float* adj       = (const float*)d_in[1];  // [8,256,256]
    const float* dep_embed = (const float*)d_in[2];  // [8,256,256,128]
    const float* weight    = (const float*)d_in[3];  // [128,128]
    const float* bias_w    = (const float*)d_in[4];  // [128]
    float* out = (float*)d_out;
    float* u   = (float*)d_ws;                       // needs 8*256*128*4 = 1 MB

    // Stage 1: HBM-bound weighted reduction over j (streams dep_embed once, NT)
    depgcn_reduce_kernel<<<Bsz * Nn, 256, 0, stream>>>(hidden, adj, dep_embed, u);

    // Stage 2: u @ W + bias via f32 WMMA (2048x128x128)
    depgcn_wmma_gemm_kernel<<<128, 256, 0, stream>>>(u, weight, bias_w, out);
}